// TransitionDown_26688926777557
// MI455X (gfx1250) — compile-verified
//
#include <hip/hip_runtime.h>
#include <hip/hip_bf16.h>
#include <stdint.h>

#define N_PTS 50000
#define IN_C  64
#define OUT_C 128
#define M_PTS 8334      // ceil(50000 / 6)
#define KNN_K 8
#define EPS   1e-5f
#define BIG   1e10f
#define XS_STRIDE 68    // 16-row LDS tile stride (pads banks, keeps 16B align)

typedef float v2f __attribute__((ext_vector_type(2)));
typedef float v4f __attribute__((ext_vector_type(4)));
typedef float v8f __attribute__((ext_vector_type(8)));

// ---------------------------------------------------------------------------
// 1) Farthest point sampling: single workgroup, min_d[50000] lives in the
//    320KB LDS (200KB), wave32 shuffle argmax + LDS tree per step.
// ---------------------------------------------------------------------------
__global__ void __launch_bounds__(1024) fps_kernel(const float* __restrict__ pos,
                                                   int* __restrict__ fps_idx) {
  extern __shared__ float smem[];
  float* min_d = smem;                         // N_PTS floats
  float* red_d = smem + N_PTS;                 // 32
  int*   red_i = (int*)(smem + N_PTS + 32);    // 32
  int*   bcast = (int*)(smem + N_PTS + 64);    // 1

  const int tid  = threadIdx.x;
  const int nthr = blockDim.x;

  for (int i = tid; i < N_PTS; i += nthr) min_d[i] = BIG;
  if (tid == 0) fps_idx[0] = 0;
  __syncthreads();

  int last = 0;
  for (int s = 1; s < M_PTS; ++s) {
    const float px = pos[last * 3 + 0];
    const float py = pos[last * 3 + 1];
    const float pz = pos[last * 3 + 2];

    float bv = -1.0f;
    int   bi = 0x7fffffff;
    for (int i = tid; i < N_PTS; i += nthr) {
      float dx = pos[i * 3 + 0] - px;
      float dy = pos[i * 3 + 1] - py;
      float dz = pos[i * 3 + 2] - pz;
      float d  = dx * dx + dy * dy + dz * dz;
      float md = fminf(min_d[i], d);
      min_d[i] = md;
      if (md > bv) { bv = md; bi = i; }   // strided ascending -> first-max kept
    }
    // wave32 argmax reduce (tie -> smaller index, matching jnp.argmax)
    #pragma unroll
    for (int off = 16; off > 0; off >>= 1) {
      float ov = __shfl_xor(bv, off, 32);
      int   oi = __shfl_xor(bi, off, 32);
      if (ov > bv || (ov == bv && oi < bi)) { bv = ov; bi = oi; }
    }
    const int wave = tid >> 5;
    if ((tid & 31) == 0) { red_d[wave] = bv; red_i[wave] = bi; }
    __syncthreads();
    if (wave == 0) {
      float v = red_d[tid & 31];
      int  ix = red_i[tid & 31];
      #pragma unroll
      for (int off = 16; off > 0; off >>= 1) {
        float ov = __shfl_xor(v, off, 32);
        int   oi = __shfl_xor(ix, off, 32);
        if (ov > v || (ov == v && oi < ix)) { v = ov; ix = oi; }
      }
      if (tid == 0) { bcast[0] = ix; fps_idx[s] = ix; }
    }
    __syncthreads();
    last = bcast[0];
  }
}

// ---------------------------------------------------------------------------
// 2) GEMM h = X @ W^T + b via V_WMMA_F32_16X16X4_F32. Block = 16 rows x 128
//    channels (8 waves, one 16-wide channel tile each). X tile staged into
//    LDS with GLOBAL_LOAD_ASYNC_TO_LDS_B128 (ASYNCcnt path, no VGPR bounce).
//    Also emits per-block per-channel (sum, sumsq) partials for GraphNorm.
// ---------------------------------------------------------------------------
__global__ void __launch_bounds__(256) gemm_kernel(const float* __restrict__ x,
                                                   const float* __restrict__ w,
                                                   const float* __restrict__ bias,
                                                   float* __restrict__ h,
                                                   float* __restrict__ psum,
                                                   float* __restrict__ psumsq) {
  __shared__ float xs[16 * XS_STRIDE];
  const int tid = threadIdx.x;
  const int m0  = blockIdx.x * 16;

  {  // async 16x64 f32 tile copy: 16B per lane, global -> LDS directly
    const int r  = tid >> 4;
    const int c4 = (tid & 15) << 2;
    const float* gsrc = x + (size_t)(m0 + r) * IN_C + c4;
    unsigned long long gaddr = (unsigned long long)(uintptr_t)gsrc;
    unsigned int laddr = (unsigned int)(uintptr_t)(&xs[r * XS_STRIDE + c4]);
    asm volatile("global_load_async_to_lds_b128 %0, %1, off"
                 :: "v"(laddr), "v"(gaddr) : "memory");
    asm volatile("s_wait_asynccnt 0x0" ::: "memory");
  }
  __syncthreads();

  const int wave = tid >> 5;
  const int lane = tid & 31;
  const int nsub = lane & 15;          // A row / B column within 16
  const int kb   = (lane >> 4) << 1;   // 0 or 2 (K sub-pair per lane-half)
  const int o0   = wave * 16;          // this wave's channel tile

  // Preload B (weights): W[o0+n][4*ks + kb + {0,1}] for all 16 K-steps.
  v2f breg[16];
  #pragma unroll
  for (int ks = 0; ks < 16; ++ks)
    breg[ks] = *(const v2f*)(w + (size_t)(o0 + nsub) * IN_C + ks * 4 + kb);

  v8f acc = {};
  #pragma unroll
  for (int ks = 0; ks < 16; ++ks) {
    v2f a = *(const v2f*)(&xs[nsub * XS_STRIDE + ks * 4 + kb]);
    acc = __builtin_amdgcn_wmma_f32_16x16x4_f32(
        false, a, false, breg[ks], (short)0, acc, false, false);
  }

  const int   ch  = o0 + nsub;
  const float bch = bias[ch];
  float s = 0.0f, sq = 0.0f;
  #pragma unroll
  for (int e = 0; e < 8; ++e) {
    float val = acc[e] + bch;
    acc[e] = val;
    s  += val;
    sq += val * val;
  }
  const int rbase = m0 + ((lane >> 4) << 3);
  #pragma unroll
  for (int e = 0; e < 8; ++e)
    h[(size_t)(rbase + e) * OUT_C + ch] = acc[e];

  // per-channel partial stats: lanes l and l^16 hold the same channel
  s  += __shfl_xor(s, 16, 32);
  sq += __shfl_xor(sq, 16, 32);
  if (lane < 16) {
    psum[(size_t)blockIdx.x * OUT_C + ch]   = s;
    psumsq[(size_t)blockIdx.x * OUT_C + ch] = sq;
  }
}

// ---------------------------------------------------------------------------
// 3) Fold partials into per-channel affine: norm(h) = h*scale + shift
// ---------------------------------------------------------------------------
__global__ void norm_stats_kernel(const float* __restrict__ psum,
                                  const float* __restrict__ psumsq,
                                  const float* __restrict__ gn_w,
                                  const float* __restrict__ gn_b,
                                  const float* __restrict__ gn_ms,
                                  float* __restrict__ scale,
                                  float* __restrict__ shift, int nblk) {
  const int c = threadIdx.x;
  if (c >= OUT_C) return;
  float s = 0.0f, sq = 0.0f;
  for (int b = 0; b < nblk; ++b) {
    s  += psum[(size_t)b * OUT_C + c];
    sq += psumsq[(size_t)b * OUT_C + c];
  }
  const float inv  = 1.0f / (float)N_PTS;
  const float mean = s * inv;
  const float ex2  = sq * inv;
  const float ms   = gn_ms[c];
  // E[(h - ms*mean)^2] = E[h^2] - 2*ms*mean^2 + ms^2*mean^2
  const float var = ex2 - 2.0f * ms * mean * mean + ms * ms * mean * mean;
  const float rs  = rsqrtf(var + EPS);
  const float sc  = rs * gn_w[c];
  scale[c] = sc;
  shift[c] = gn_b[c] - ms * mean * sc;
}

// ---------------------------------------------------------------------------
// 4) KNN: one wave32 per center. Per-lane sorted top-8 in registers over a
//    strided scan of N, then 5-level bitonic shfl_xor merge across the wave.
// ---------------------------------------------------------------------------
__device__ __forceinline__ void ceex(float& d0, int& i0, float& d1, int& i1) {
  bool sw = (d1 < d0) || (d1 == d0 && i1 < i0);
  float td = sw ? d1 : d0;  float tD = sw ? d0 : d1;
  int   ti = sw ? i1 : i0;  int   tI = sw ? i0 : i1;
  d0 = td; d1 = tD; i0 = ti; i1 = tI;
}

__global__ void __launch_bounds__(256) knn_kernel(const float* __restrict__ pos,
                                                  const int* __restrict__ fps_idx,
                                                  int* __restrict__ col) {
  const int wave = threadIdx.x >> 5;
  const int lane = threadIdx.x & 31;
  const int m = blockIdx.x * 8 + wave;
  if (m >= M_PTS) return;

  const int   ci = fps_idx[m];
  const float cx = pos[ci * 3 + 0];
  const float cy = pos[ci * 3 + 1];
  const float cz = pos[ci * 3 + 2];

  float bd[KNN_K];
  int   bi[KNN_K];
  #pragma unroll
  for (int j = 0; j < KNN_K; ++j) { bd[j] = 3.0e38f; bi[j] = 0x7fffffff; }

  for (int i = lane; i < N_PTS; i += 32) {
    float dx = pos[i * 3 + 0] - cx;
    float dy = pos[i * 3 + 1] - cy;
    float dz = pos[i * 3 + 2] - cz;
    float d  = dx * dx + dy * dy + dz * dz;
    if (d < bd[KNN_K - 1] || (d == bd[KNN_K - 1] && i < bi[KNN_K - 1])) {
      float cd = d; int cix = i;   // bubble-insert into sorted ascending list
      #pragma unroll
      for (int j = 0; j < KNN_K; ++j) {
        bool sw = (cd < bd[j]) || (cd == bd[j] && cix < bi[j]);
        float td = sw ? bd[j] : cd;  int ti = sw ? bi[j] : cix;
        bd[j] = sw ? cd : bd[j];     bi[j] = sw ? cix : bi[j];
        cd = td; cix = ti;
      }
    }
  }

  // cross-lane merge: both partners compute the identical merged top-8
  #pragma unroll
  for (int off = 1; off < 32; off <<= 1) {
    float od[KNN_K]; int oi[KNN_K];
    #pragma unroll
    for (int j = 0; j < KNN_K; ++j) {
      od[j] = __shfl_xor(bd[j], off, 32);
      oi[j] = __shfl_xor(bi[j], off, 32);
    }
    float nd[KNN_K]; int ni[KNN_K];
    #pragma unroll
    for (int j = 0; j < KNN_K; ++j) {           // 8 smallest of the union
      float a = bd[j], b = od[KNN_K - 1 - j];
      int  ai = bi[j], ob = oi[KNN_K - 1 - j];
      bool tk = (a < b) || (a == b && ai < ob);
      nd[j] = tk ? a : b;
      ni[j] = tk ? ai : ob;
    }
    // bitonic sort-8 (stages 4, 2, 1)
    ceex(nd[0], ni[0], nd[4], ni[4]); ceex(nd[1], ni[1], nd[5], ni[5]);
    ceex(nd[2], ni[2], nd[6], ni[6]); ceex(nd[3], ni[3], nd[7], ni[7]);
    ceex(nd[0], ni[0], nd[2], ni[2]); ceex(nd[1], ni[1], nd[3], ni[3]);
    ceex(nd[4], ni[4], nd[6], ni[6]); ceex(nd[5], ni[5], nd[7], ni[7]);
    ceex(nd[0], ni[0], nd[1], ni[1]); ceex(nd[2], ni[2], nd[3], ni[3]);
    ceex(nd[4], ni[4], nd[5], ni[5]); ceex(nd[6], ni[6], nd[7], ni[7]);
    #pragma unroll
    for (int j = 0; j < KNN_K; ++j) { bd[j] = nd[j]; bi[j] = ni[j]; }
  }

  if (lane == 0) {
    #pragma unroll
    for (int j = 0; j < KNN_K; ++j) col[(size_t)m * KNN_K + j] = bi[j];
  }
}

// ---------------------------------------------------------------------------
// 5) Gather + affine norm + ReLU + max over K neighbors
// ---------------------------------------------------------------------------
__global__ void __launch_bounds__(128) gather_kernel(const float* __restrict__ h,
                                                     const int* __restrict__ col,
                                                     const float* __restrict__ scale,
                                                     const float* __restrict__ shift,
                                                     float* __restrict__ out) {
  const int m = blockIdx.x;
  const int c = threadIdx.x;
  const float sc = scale[c], sh = shift[c];
  float best = -3.0e38f;
  #pragma unroll
  for (int k = 0; k < KNN_K; ++k) {
    int idx = col[(size_t)m * KNN_K + k];
    float v = fmaf(h[(size_t)idx * OUT_C + c], sc, sh);
    best = fmaxf(best, v);
  }
  out[(size_t)m * OUT_C + c] = fmaxf(best, 0.0f);
}

// ---------------------------------------------------------------------------
// 6) Tail of d_out: sub_pos = pos[fps_idx], then zeros (sub_batch == 0)
// ---------------------------------------------------------------------------
__global__ void finalize_kernel(const float* __restrict__ pos,
                                const int* __restrict__ fps_idx,
                                float* __restrict__ tail, int tail_elems) {
  int t = blockIdx.x * blockDim.x + threadIdx.x;
  if (t >= tail_elems) return;
  if (t < M_PTS * 3) {
    int m = t / 3, d = t - m * 3;
    tail[t] = pos[(size_t)fps_idx[m] * 3 + d];
  } else {
    tail[t] = 0.0f;
  }
}

// ---------------------------------------------------------------------------
extern "C" void kernel_launch(void* const* d_in, const int* in_sizes, int n_in,
                              void* d_out, int out_size, void* d_ws, size_t ws_size,
                              hipStream_t stream) {
  const float* x     = (const float*)d_in[0];
  const float* pos   = (const float*)d_in[1];
  /* d_in[2] = batch (all zeros, unused) */
  const float* lin_w = (const float*)d_in[3];
  const float* lin_b = (const float*)d_in[4];
  const float* gn_w  = (const float*)d_in[5];
  const float* gn_b  = (const float*)d_in[6];
  const float* gn_ms = (const float*)d_in[7];

  char* ws = (char*)d_ws;
  size_t off = 0;
  auto alloc = [&](size_t bytes) -> void* {
    void* p = ws + off;
    off = (off + bytes + 255) & ~(size_t)255;
    return p;
  };
  int*   fps_idx = (int*)alloc((size_t)M_PTS * sizeof(int));
  int*   col     = (int*)alloc((size_t)M_PTS * KNN_K * sizeof(int));
  float* h       = (float*)alloc((size_t)N_PTS * OUT_C * sizeof(float));
  const int nblk = N_PTS / 16;  // 3125 (50000 is a multiple of 16)
  float* psum    = (float*)alloc((size_t)nblk * OUT_C * sizeof(float));
  float* psumsq  = (float*)alloc((size_t)nblk * OUT_C * sizeof(float));
  float* scale   = (float*)alloc(OUT_C * sizeof(float));
  float* shift   = (float*)alloc(OUT_C * sizeof(float));

  // 1) FPS with 200KB of dynamic LDS (min_d resident in the 320KB WGP LDS)
  const size_t smem = (size_t)(N_PTS + 96) * sizeof(float);
  (void)hipFuncSetAttribute((const void*)fps_kernel,
                            hipFuncAttributeMaxDynamicSharedMemorySize, (int)smem);
  fps_kernel<<<1, 1024, smem, stream>>>(pos, fps_idx);

  // 2) WMMA GEMM + GraphNorm partials (async global->LDS staging)
  gemm_kernel<<<nblk, 256, 0, stream>>>(x, lin_w, lin_b, h, psum, psumsq);

  // 3) Per-channel scale/shift (deterministic two-stage reduction)
  norm_stats_kernel<<<1, OUT_C, 0, stream>>>(psum, psumsq, gn_w, gn_b, gn_ms,
                                             scale, shift, nblk);

  // 4) KNN (one wave32 per center)
  knn_kernel<<<(M_PTS + 7) / 8, 256, 0, stream>>>(pos, fps_idx, col);

  // 5) Gather-max into the first M*128 elements of d_out
  gather_kernel<<<M_PTS, OUT_C, 0, stream>>>(h, col, scale, shift, (float*)d_out);

  // 6) sub_pos + zero sub_batch tail
  const int tail = out_size - M_PTS * OUT_C;
  if (tail > 0) {
    float* tp = (float*)d_out + (size_t)M_PTS * OUT_C;
    finalize_kernel<<<(tail + 255) / 256, 256, 0, stream>>>(pos, fps_idx, tp, tail);
  }
}